// BehaviorFluidFlow_43748536877186
// MI455X (gfx1250) — compile-verified
//
#include <hip/hip_runtime.h>
#include <cstdint>
#include <cstddef>

// Problem constants from the reference
#define BB 16
#define CC 20
#define HH 512
#define WW 512
#define HW (HH * WW)          // 262144
#define CHW (CC * HW)         // 5242880

#define AS1 __attribute__((address_space(1)))
#define AS3 __attribute__((address_space(3)))

#if defined(__HIP_DEVICE_COMPILE__) && defined(__gfx1250__) &&                     \
    __has_builtin(__builtin_amdgcn_global_load_async_to_lds_b128) &&               \
    __has_builtin(__builtin_amdgcn_global_store_async_from_lds_b128) &&            \
    __has_builtin(__builtin_amdgcn_s_wait_asynccnt)
#define USE_ASYNC_LDS 1
#else
#define USE_ASYNC_LDS 0
#endif

typedef float float4v __attribute__((ext_vector_type(4)));
typedef int   v4i     __attribute__((ext_vector_type(4)));

// One 512-thread workgroup per (b, h) row. Thread = column w.
// Row state (20 channels x 512 columns, 40KB) lives in LDS for all 10 CA steps.
__global__ __launch_bounds__(512) void fluid_flow_kernel(
    const float* __restrict__ world_in,
    const float* __restrict__ rand_movement,
    const float* __restrict__ did_gravity,
    float* __restrict__ world_out)
{
    __shared__ float lds[CC * WW];   // 40 KB: channel-major row tile
    __shared__ float blArr[WW];      // 2 KB: becomes_left mask exchange

    const int w  = threadIdx.x;          // column 0..511
    const int bh = blockIdx.x;           // 0..8191
    const int b  = bh >> 9;
    const int h  = bh & (HH - 1);
    const size_t rowBase = (size_t)b * CHW + (size_t)h * WW;   // &world[b][0][h][0]

    // ---- Load 20x512 row tile into LDS (2560 x 16B chunks, 5 per thread) ----
#if USE_ASYNC_LDS
    #pragma unroll
    for (int k = 0; k < 5; ++k) {
        const int g  = w + k * WW;       // chunk id 0..2559
        const int c  = g >> 7;           // channel = chunk / 128
        const int w4 = g & 127;          // float4 index within channel row
        const float* src = world_in + rowBase + (size_t)c * HW + (size_t)w4 * 4;
        float* dst = &lds[c * WW + w4 * 4];
        __builtin_amdgcn_global_load_async_to_lds_b128(
            (AS1 v4i*)src, (AS3 v4i*)dst, 0, 0);
    }
    __builtin_amdgcn_s_wait_asynccnt(0);
#else
    #pragma unroll
    for (int k = 0; k < 5; ++k) {
        const int g  = w + k * WW;
        const int c  = g >> 7;
        const int w4 = g & 127;
        const float4v v =
            *(const float4v*)(world_in + rowBase + (size_t)c * HW + (size_t)w4 * 4);
        *(float4v*)&lds[c * WW + w4 * 4] = v;
    }
#endif
    __syncthreads();

    const int   pix = b * HW + h * WW + w;
    const float rm  = rand_movement[pix];
    const bool  ndg = (did_gravity[pix] <= 0.0f);   // not_did_gravity
    float nf = 0.0f;                                 // new_fm accumulator

    const int elems[5] = {0, 3, 4, 10, 11};

    #pragma unroll
    for (int ei = 0; ei < 5; ++ei) {
        const int eoff = elems[ei] * WW;
        #pragma unroll
        for (int d = 0; d < 2; ++d) {
            const bool fall_left = (d == 0);
            const int  sh = fall_left ? 1 : -1;
            const int  li = (w - sh) & (WW - 1);   // index of gdir() source
            const int  ri = (w + sh) & (WW - 1);   // index of gnot() source

            __syncthreads();  // previous step's channel writes visible

            // becomes_left for this pixel (b[w] == becomes_left[w+sh], via blArr)
            const float fmv      = lds[16 * WW + w];
            const bool  fall_dir = ((rm + fmv) + nf) > 0.5f;
            const bool  matching = fall_left ? fall_dir : !fall_dir;
            const bool  is_el    = (lds[eoff + w] == 1.0f);
            const float dw       = lds[14 * WW + w];
            const float dl       = lds[14 * WW + li];
            const bool  gw       = (lds[15 * WW + w]  == 1.0f);
            const bool  gl       = (lds[15 * WW + li] == 1.0f);
            const bool  bl = matching && is_el && ndg &&
                             ((dw - dl) > 0.0f) && gw && gl;
            const float a = bl ? 1.0f : 0.0f;
            blArr[w] = a;
            __syncthreads();  // masks visible

            const float bb = blArr[ri];  // becomes_right at this pixel
            nf += bb * (fall_left ? 2.0f : -2.0f);

            const bool active = (a != 0.0f) | (bb != 0.0f);
            float vals[CC];
            if (active) {
                const float keep = (1.0f - a) * (1.0f - bb);
                #pragma unroll
                for (int c = 0; c < CC; ++c) {
                    vals[c] = keep * lds[c * WW + w]
                            + a    * lds[c * WW + li]
                            + bb   * lds[c * WW + ri];
                }
            }
            __syncthreads();  // all reads of current state done
            if (active) {
                #pragma unroll
                for (int c = 0; c < CC; ++c) lds[c * WW + w] = vals[c];
            }
        }
    }

    __syncthreads();
    lds[16 * WW + w] = nf;   // world[:, fm_c] = new_fm
    __syncthreads();

    // ---- Store 20x512 row tile to world_out ----
#if USE_ASYNC_LDS
    #pragma unroll
    for (int k = 0; k < 5; ++k) {
        const int g  = w + k * WW;
        const int c  = g >> 7;
        const int w4 = g & 127;
        float* dst = world_out + rowBase + (size_t)c * HW + (size_t)w4 * 4;
        __builtin_amdgcn_global_store_async_from_lds_b128(
            (AS1 v4i*)dst, (AS3 v4i*)&lds[c * WW + w4 * 4], 0, 0);
    }
    __builtin_amdgcn_s_wait_asynccnt(0);
#else
    #pragma unroll
    for (int k = 0; k < 5; ++k) {
        const int g  = w + k * WW;
        const int c  = g >> 7;
        const int w4 = g & 127;
        const float4v v = *(const float4v*)&lds[c * WW + w4 * 4];
        *(float4v*)(world_out + rowBase + (size_t)c * HW + (size_t)w4 * 4) = v;
    }
#endif
}

extern "C" void kernel_launch(void* const* d_in, const int* in_sizes, int n_in,
                              void* d_out, int out_size, void* d_ws, size_t ws_size,
                              hipStream_t stream) {
    (void)d_ws; (void)ws_size; (void)out_size; (void)n_in;

    const float* world = (const float*)d_in[0];
    const float* rm    = (const float*)d_in[1];
    const float* dg    = (const float*)d_in[5];
    float* out = (float*)d_out;

    // world output occupies the first B*C*H*W floats of d_out
    fluid_flow_kernel<<<BB * HH, WW, 0, stream>>>(world, rm, dg, out);

    // Pass-through outputs: rand_movement, rand_interact, rand_element,
    // velocity_field, did_gravity — concatenated flat after world.
    size_t off = (size_t)BB * CHW;
    for (int i = 1; i < 6; ++i) {
        const size_t n = (size_t)in_sizes[i];
        (void)hipMemcpyAsync(out + off, d_in[i], n * sizeof(float),
                             hipMemcpyDeviceToDevice, stream);
        off += n;
    }
}